// SilhouetteShader_68848325755333
// MI455X (gfx1250) — compile-verified
//
#include <hip/hip_runtime.h>
#include <cstdint>

typedef float v4f __attribute__((ext_vector_type(4)));
typedef int   v4i __attribute__((ext_vector_type(4)));

// LDS staging: 64B of dists per lane, padded to 80B so ds_load_b128 phases
// are bank-conflict-free (20*t mod 64 -> distinct multiples of 4 per 16 lanes).
#define LDS_STRIDE 80
#define BLOCK 256

// exp(-d * 1e4) == exp2(d * -1e4 * log2(e))
#define NEG_SCALE_LOG2E (-14426.950408889634f)

__global__ __launch_bounds__(BLOCK) void silhouette_kernel(
    const float* __restrict__ colors,       // [npix, 16, 3] f32
    const float* __restrict__ dists,        // [npix, 16]    f32
    const long long* __restrict__ p2f,      // [npix, 16]    i64
    float* __restrict__ out,                // [npix, 4]     f32
    unsigned npix)
{
    __shared__ char smem[BLOCK * LDS_STRIDE];

    const unsigned t   = threadIdx.x;
    const unsigned pix = blockIdx.x * BLOCK + t;
    if (pix >= npix) return;

    // ---- 1) kick off async DMA of this lane's 16 dists (64B) into LDS ----
    const char* gsrc   = (const char*)(dists + (size_t)pix * 16);
    char*       lptr   = smem + t * LDS_STRIDE;
    uint32_t    lds_lo = (uint32_t)(uintptr_t)lptr;   // LDS byte address = low 32 bits of the pointer

#define ASYNC_B128(OFFS) \
    asm volatile("global_load_async_to_lds_b128 %0, %1, off offset:" OFFS \
                 :: "v"(lds_lo), "v"(gsrc) : "memory")
    ASYNC_B128("0");
    ASYNC_B128("16");
    ASYNC_B128("32");
    ASYNC_B128("48");
#undef ASYNC_B128

    // ---- 2) overlap: pix_to_face masks (int64 sign bit lives in the high dword) ----
    const v4i* pq = (const v4i*)(p2f + (size_t)pix * 16);   // 8 x b128, NT single-use stream
    int hi[16];
#pragma unroll
    for (int j = 0; j < 8; ++j) {
        v4i q = __builtin_nontemporal_load(pq + j); // {lo0,hi0,lo1,hi1}
        hi[2 * j]     = q.y;
        hi[2 * j + 1] = q.w;
    }

    // ---- 3) overlap: colors[pix, 0, 0:3] (row is 192B; grab aligned 16B) ----
    v4f c = __builtin_nontemporal_load((const v4f*)(colors + (size_t)pix * 48));

    // ---- 4) wait for the async DMA, then consume dists from LDS ----
    asm volatile("s_wait_asynccnt 0" ::: "memory");

    // prod_k sigmoid(d_k/sigma) = 1 / prod_k (1 + exp(-d_k * 1e4)) over masked k
    float denom = 1.0f;
#pragma unroll
    for (int j = 0; j < 4; ++j) {
        v4f d = *(const v4f*)(lptr + j * 16);       // ds_load_b128, conflict-free
#pragma unroll
        for (int e = 0; e < 4; ++e) {
            const int k = j * 4 + e;
#if __has_builtin(__builtin_amdgcn_exp2f)
            float ex = __builtin_amdgcn_exp2f(d[e] * NEG_SCALE_LOG2E); // v_mul + v_exp_f32
#else
            float ex = __expf(d[e] * -10000.0f);
#endif
            denom *= (hi[k] >= 0) ? (1.0f + ex) : 1.0f;
        }
    }

    const float alpha = 1.0f - __builtin_amdgcn_rcpf(denom);

    v4f r;
    r.x = c.x; r.y = c.y; r.z = c.z; r.w = alpha;
    __builtin_nontemporal_store(r, (v4f*)(out + (size_t)pix * 4)); // b128 NT store
}

extern "C" void kernel_launch(void* const* d_in, const int* in_sizes, int n_in,
                              void* d_out, int out_size, void* d_ws, size_t ws_size,
                              hipStream_t stream) {
    (void)n_in; (void)out_size; (void)d_ws; (void)ws_size;
    const float*     colors = (const float*)d_in[0];
    const float*     dists  = (const float*)d_in[1];
    const long long* p2f    = (const long long*)d_in[2];
    float*           out    = (float*)d_out;

    const unsigned npix   = (unsigned)(in_sizes[1] / 16);   // dists is [npix,16]
    const unsigned blocks = (npix + BLOCK - 1) / BLOCK;

    hipLaunchKernelGGL(silhouette_kernel, dim3(blocks), dim3(BLOCK), 0, stream,
                       colors, dists, p2f, out, npix);
}